// MONSTOR_72954314490496
// MI455X (gfx1250) — compile-verified
//
#include <hip/hip_runtime.h>

#define NN 100000
#define NE 1600000

typedef __attribute__((ext_vector_type(2))) float v2f;
typedef __attribute__((ext_vector_type(8))) float v8f;

// Order-preserving float -> uint mapping so segment-max can use native
// non-returning GLOBAL_ATOMIC_MAX_U32 (values may be negative; 0u acts as -inf).
__device__ __forceinline__ unsigned mapf(float f) {
  unsigned u = __float_as_uint(f);
  return (u & 0x80000000u) ? ~u : (u | 0x80000000u);
}
__device__ __forceinline__ float unmapf(unsigned u) {
  unsigned b = (u & 0x80000000u) ? (u ^ 0x80000000u) : ~u;
  return __uint_as_float(b);
}

__global__ void fill_u32_kernel(unsigned* __restrict__ p, unsigned v, long long n) {
  long long i = (long long)blockIdx.x * blockDim.x + threadIdx.x;
  if (i < n) p[i] = v;
}

// deg = segment_sum(1, dst); dub = segment_sum(features[src,14] * ew, dst)
__global__ void edge_pre_kernel(const int* __restrict__ src, const int* __restrict__ dst,
                                const float* __restrict__ ew, const float* __restrict__ feat,
                                int* __restrict__ deg, float* __restrict__ dub, int nE) {
  int e = blockIdx.x * blockDim.x + threadIdx.x;
  if (e >= nE) return;
  int s = src[e], d = dst[e];
  atomicAdd(&deg[d], 1);
  atomicAdd(&dub[d], feat[(long long)s * 16 + 14] * ew[e]);
}

// segment_max(h[src]*ew, dst) via mapped-u32 atomic max; 8 lanes per edge.
__global__ void edge_max_kernel(const int* __restrict__ src, const int* __restrict__ dst,
                                const float* __restrict__ ew, const float* __restrict__ h,
                                unsigned* __restrict__ neigh, int d, int nE) {
  long long t = (long long)blockIdx.x * blockDim.x + threadIdx.x;
  int e  = (int)(t >> 3);
  int c0 = (int)(t & 7);
  if (e >= nE) return;
  int s = src[e], dd = dst[e];
  float w = ew[e];
  const float* hr = h + (long long)s * d;
  unsigned* nr = neigh + (long long)dd * d;
  for (int c = c0; c < d; c += 8)
    atomicMax(&nr[c], mapf(hr[c] * w));
}

// In-place decode: neigh(float) = deg>0 ? unmap(u) : 0
__global__ void decode_kernel(unsigned* __restrict__ buf, const int* __restrict__ deg,
                              int d, long long total) {
  long long i = (long long)blockIdx.x * blockDim.x + threadIdx.x;
  if (i >= total) return;
  int row = (int)(i / d);
  float f = (deg[row] > 0) ? unmapf(buf[i]) : 0.0f;
  ((float*)buf)[i] = f;
}

// Generic fp32 WMMA GEMM: Y[N x dOut] = A[N x Kt] * W[Kt x dOut] + bias (opt relu).
// A is split across two sources: k < ksplit from A0, else from A1 (for the
// concat(feat, neigh) layers without materializing the concat).
// One 16x16 output tile per wave; K-loop of V_WMMA_F32_16X16X4_F32.
// W is staged in LDS (tiny: <= 96x48 f32).
__global__ void gemm_conv_kernel(const float* __restrict__ A0, int lda0,
                                 const float* __restrict__ A1, int lda1, int ksplit,
                                 const float* __restrict__ W, const float* __restrict__ bias,
                                 float* __restrict__ Y, int ldy,
                                 int nRows, int Kt, int dOut, int relu) {
  extern __shared__ float sW[];   // Kt * dOut
  int tid = threadIdx.x;
  for (int i = tid; i < Kt * dOut; i += blockDim.x) sW[i] = W[i];
  __syncthreads();

  int wid  = tid >> 5;
  int lane = tid & 31;
  int tilesN = dOut >> 4;
  int nTiles = (nRows >> 4) * tilesN;
  int tileId = blockIdx.x * (blockDim.x >> 5) + wid;
  if (tileId >= nTiles) return;          // whole-wave uniform -> EXEC stays all-ones
  int m0 = (tileId / tilesN) << 4;
  int n0 = (tileId % tilesN) << 4;

  int lm = lane & 15;            // A row within tile / B,D column within tile
  int kh = (lane >> 4) << 1;     // K sub-offset for upper/lower half-wave: 0 or 2

  v8f c = {};
  for (int k0 = 0; k0 < Kt; k0 += 4) {
    int k = k0 + kh;
    // A fragment: lane holds A[m0+lm][k], A[m0+lm][k+1]  (16x4 f32 layout)
    v2f a;
    if (k0 < ksplit) {
      const float* p = A0 + (long long)(m0 + lm) * lda0 + k;
      a.x = p[0]; a.y = p[1];
    } else {
      const float* p = A1 + (long long)(m0 + lm) * lda1 + (k - ksplit);
      a.x = p[0]; a.y = p[1];
    }
    // B fragment: lane holds W[k][n0+lm], W[k+1][n0+lm]  (4x16 f32 layout)
    v2f b;
    b.x = sW[(long long)k * dOut + n0 + lm];
    b.y = sW[(long long)(k + 1) * dOut + n0 + lm];
    c = __builtin_amdgcn_wmma_f32_16x16x4_f32(false, a, false, b, (short)0, c,
                                              false, false);
  }

  // D layout: VGPR r -> row (r + 8*(lane>>4)), col (lane&15)
  int mo = (lane >> 4) << 3;
  float bn = bias[n0 + lm];
#pragma unroll
  for (int r = 0; r < 8; ++r) {
    float v = c[r] + bn;
    if (relu) v = fmaxf(v, 0.0f);
    Y[(long long)(m0 + mo + r) * ldy + (n0 + lm)] = v;
  }
}

// Layer-2 node conv (d_out = 1, too skinny for WMMA) fused with the final
// upper-bound clamp: out = min(now + relu(concat@Wn2+bn2), clip(now+dub,0,1))
__global__ void proj_out_kernel(const float* __restrict__ feat2, const float* __restrict__ neighf,
                                const float* __restrict__ Wn2, const float* __restrict__ bn2,
                                const float* __restrict__ features, const float* __restrict__ dub,
                                float* __restrict__ out, int N) {
  int n = blockIdx.x * blockDim.x + threadIdx.x;
  if (n >= N) return;
  float acc = bn2[0];
  const float* f = feat2 + (long long)n * 48;
  const float* g = neighf + (long long)n * 48;
#pragma unroll
  for (int k = 0; k < 48; ++k) acc += f[k] * Wn2[k];
#pragma unroll
  for (int k = 0; k < 48; ++k) acc += g[k] * Wn2[48 + k];
  acc = fmaxf(acc, 0.0f);
  float now = features[(long long)n * 16 + 15];
  float ub = fminf(fmaxf(now + dub[n], 0.0f), 1.0f);
  out[n] = fminf(now + acc, ub);
}

extern "C" void kernel_launch(void* const* d_in, const int* in_sizes, int n_in,
                              void* d_out, int out_size, void* d_ws, size_t ws_size,
                              hipStream_t stream) {
  const float* features = (const float*)d_in[0];
  const float* ew  = (const float*)d_in[1];
  const int*   src = (const int*)d_in[2];
  const int*   dst = (const int*)d_in[3];
  const float* Wp0 = (const float*)d_in[4];
  const float* bp0 = (const float*)d_in[5];
  const float* Wn0 = (const float*)d_in[6];
  const float* bn0 = (const float*)d_in[7];
  const float* Wp1 = (const float*)d_in[8];
  const float* bp1 = (const float*)d_in[9];
  const float* Wn1 = (const float*)d_in[10];
  const float* bn1 = (const float*)d_in[11];
  const float* Wp2 = (const float*)d_in[12];
  const float* bp2 = (const float*)d_in[13];
  const float* Wn2 = (const float*)d_in[14];
  const float* bn2 = (const float*)d_in[15];

  // Workspace layout (~58.4 MB total)
  char* ws = (char*)d_ws;
  float*    bufA  = (float*)(ws);                         // N*48 f32
  float*    bufB  = (float*)(ws + 19200000);              // N*48 f32
  unsigned* neigh = (unsigned*)(ws + 38400000);           // N*48 u32/f32
  int*      deg   = (int*)(ws + 57600000);                // N i32
  float*    dub   = (float*)(ws + 58000000);              // N f32

  const int T = 256;
  auto gemm = [&](const float* A0, int lda0, const float* A1, int lda1, int ksplit,
                  const float* W, const float* bias, float* Y, int ldy,
                  int Kt, int dOut, int relu) {
    int nTiles = (NN / 16) * (dOut / 16);
    int blocks = (nTiles + 7) / 8;             // 8 waves (tiles) per 256-thr block
    size_t shm = (size_t)Kt * dOut * sizeof(float);
    gemm_conv_kernel<<<blocks, T, shm, stream>>>(A0, lda0, A1, lda1, ksplit,
                                                 W, bias, Y, ldy, NN, Kt, dOut, relu);
  };
  auto fill = [&](unsigned* p, long long n) {
    fill_u32_kernel<<<(int)((n + T - 1) / T), T, 0, stream>>>(p, 0u, n);
  };
  auto emax = [&](const float* h, int d) {
    long long tt = (long long)NE * 8;
    edge_max_kernel<<<(int)((tt + T - 1) / T), T, 0, stream>>>(src, dst, ew, h, neigh, d, NE);
  };
  auto decode = [&](int d) {
    long long tt = (long long)NN * d;
    decode_kernel<<<(int)((tt + T - 1) / T), T, 0, stream>>>(neigh, deg, d, tt);
  };

  // degree + delta_ub (deg and dub are adjacent -> one zero-fill)
  fill((unsigned*)deg, 2LL * NN);
  edge_pre_kernel<<<(NE + T - 1) / T, T, 0, stream>>>(src, dst, ew, features, deg, dub, NE);

  // ---- Layer 0 (d_in=16, d_out=48) ----
  gemm(features, 16, features, 16, 16, Wp0, bp0, bufA, 16, 16, 16, 0);   // h0 -> bufA
  fill(neigh, (long long)NN * 16);
  emax(bufA, 16);
  decode(16);
  gemm(features, 16, (const float*)neigh, 16, 16, Wn0, bn0, bufB, 48, 32, 48, 1); // feat1 -> bufB

  // ---- Layer 1 (d_in=48, d_out=48) ----
  gemm(bufB, 48, bufB, 48, 48, Wp1, bp1, bufA, 48, 48, 48, 0);           // h1 -> bufA
  fill(neigh, (long long)NN * 48);
  emax(bufA, 48);
  decode(48);
  gemm(bufB, 48, (const float*)neigh, 48, 48, Wn1, bn1, bufA, 48, 96, 48, 1); // feat2 -> bufA

  // ---- Layer 2 (d_in=48, d_out=1) + final clamp ----
  gemm(bufA, 48, bufA, 48, 48, Wp2, bp2, bufB, 48, 48, 48, 0);           // h2 -> bufB
  fill(neigh, (long long)NN * 48);
  emax(bufB, 48);
  decode(48);
  proj_out_kernel<<<(NN + T - 1) / T, T, 0, stream>>>(bufA, (const float*)neigh,
                                                      Wn2, bn2, features, dub,
                                                      (float*)d_out, NN);
}